// QuantumDilatedCNN_40707700032199
// MI455X (gfx1250) — compile-verified
//
#include <hip/hip_runtime.h>
#include <cmath>

// ---------- complex helpers ----------
__device__ __forceinline__ float2 cmul(float2 a, float2 b){
    return make_float2(a.x*b.x - a.y*b.y, a.x*b.y + a.y*b.x);
}
__device__ __forceinline__ float2 cadd(float2 a, float2 b){
    return make_float2(a.x+b.x, a.y+b.y);
}

// ---------- static gate program (derived from the reference circuit) ----------
// types: 0=U3(1q, embedded as I(wa) (x) U(wb)), 1=ZZ, 2=YY, 3=XX, 4=CU3(ctrl=wa,tgt=wb),
//        5=PAULI rotation exp(-i th/2 * P_a (x) P_b), aux = pa*4+pb
struct GD { int type, wa, wb, aux, poff; };

__constant__ GD g_gates[66] = {
    // ---- layer 0 conv, pair (0,2) ----
    {0,1,0,0,  0},{0,3,2,0, 33},{1,0,2,0,  6},{2,0,2,0,  7},{3,0,2,0,  8},{0,1,0,0,  9},{0,3,2,0, 42},
    // pair (1,3)
    {0,0,1,0, 15},{0,2,3,0, 48},{1,1,3,0, 21},{2,1,3,0, 22},{3,1,3,0, 23},{0,0,1,0, 24},{0,2,3,0, 57},
    // pair (4,6)
    {0,5,4,0, 60},{0,7,6,0, 93},{1,4,6,0, 66},{2,4,6,0, 67},{3,4,6,0, 68},{0,5,4,0, 69},{0,7,6,0,102},
    // pair (5,7)
    {0,4,5,0, 75},{0,6,7,0,108},{1,5,7,0, 81},{2,5,7,0, 82},{3,5,7,0, 83},{0,4,5,0, 84},{0,6,7,0,117},
    // ---- pool 0 ----
    {4,1,0,0,  0},{4,3,2,0,  3},{4,5,4,0,  6},{4,7,6,0,  9},
    // ---- layer 1 conv, pair (0,2) (conv base 120) ----
    {0,1,0,0,120},{0,3,2,0,153},{1,0,2,0,126},{2,0,2,0,127},{3,0,2,0,128},{0,1,0,0,129},{0,3,2,0,162},
    // leftover singles w=4
    {0,5,4,0,180},{0,5,4,0,183},{0,5,4,0,186},{0,5,4,0,189},{0,5,4,0,192},
    // leftover singles w=6
    {0,7,6,0,210},{0,7,6,0,213},{0,7,6,0,216},{0,7,6,0,219},{0,7,6,0,222},
    // ---- pool 1 (pool base 12) ----
    {4,2,0,0, 12},{4,6,4,0, 15},
    // ---- arbitrary_unitary on wires (0,4): 15 Gray-code Pauli words ----
    {5,0,4, 4, 0},{5,0,4, 8, 1},{5,0,4,12, 2},{5,0,4,13, 3},{5,0,4, 1, 4},
    {5,0,4, 5, 5},{5,0,4, 9, 6},{5,0,4,10, 7},{5,0,4,14, 8},{5,0,4, 2, 9},
    {5,0,4, 6,10},{5,0,4, 7,11},{5,0,4,11,12},{5,0,4,15,13},{5,0,4, 3,14}
};

__device__ __forceinline__ float2 pel(int p, int r, int c){
    // Pauli matrix element P[p][r][c]
    if (p == 0) return (r==c) ? make_float2(1.f,0.f) : make_float2(0.f,0.f);
    if (p == 1) return (r!=c) ? make_float2(1.f,0.f) : make_float2(0.f,0.f);
    if (p == 2) return (r==c) ? make_float2(0.f,0.f)
                              : (r==0 ? make_float2(0.f,-1.f) : make_float2(0.f,1.f));
    return (r==c) ? (r==0 ? make_float2(1.f,0.f) : make_float2(-1.f,0.f)) : make_float2(0.f,0.f);
}

// ---------- kernel A: reduced = tanh(x @ fcw^T + b); store cos/sin of theta/2 ----------
__global__ void __launch_bounds__(256) fc_tanh(const float* __restrict__ x,
                                               const float* __restrict__ fcw,
                                               const float* __restrict__ fcb,
                                               float2* __restrict__ cs){
    int wave = threadIdx.x >> 5;
    int lane = threadIdx.x & 31;
    int row  = blockIdx.x * 8 + wave;               // 2048 blocks * 8 waves = 16384 rows
    const float4* xr = (const float4*)(x + (size_t)row * 3072);
    float acc[8];
#pragma unroll
    for (int w = 0; w < 8; ++w) acc[w] = 0.f;
    for (int it = 0; it < 24; ++it){                // 24*32 = 768 float4 = 3072 floats
        int k4 = lane + it * 32;
        float4 xv = xr[k4];
#pragma unroll
        for (int w = 0; w < 8; ++w){
            float4 wv = ((const float4*)(fcw + (size_t)w * 3072))[k4];
            acc[w] += xv.x*wv.x + xv.y*wv.y + xv.z*wv.z + xv.w*wv.w;
        }
    }
#pragma unroll
    for (int w = 0; w < 8; ++w){
#pragma unroll
        for (int m = 16; m >= 1; m >>= 1) acc[w] += __shfl_xor(acc[w], m, 32);
    }
    if (lane == 0){
#pragma unroll
        for (int w = 0; w < 8; ++w){
            float half_t = tanhf(acc[w] + fcb[w]) * 0.5f;
            cs[(size_t)row * 8 + w] = make_float2(cosf(half_t), sinf(half_t));
        }
    }
}

// ---------- kernel B: build 66 gate matrices (4x4 complex each) ----------
__global__ void build_gates(const float* __restrict__ conv, const float* __restrict__ pool,
                            const float* __restrict__ last, float2* __restrict__ gws){
    int g = threadIdx.x;
    if (g >= 66) return;
    GD d = g_gates[g];
    float2 M[16];
#pragma unroll
    for (int e = 0; e < 16; ++e) M[e] = make_float2(0.f, 0.f);

    if (d.type == 0 || d.type == 4){
        const float* P = (d.type == 0) ? (conv + d.poff) : (pool + d.poff);
        float th = P[0], ph = P[1], lam = P[2];
        float c = cosf(th * 0.5f), s = sinf(th * 0.5f);
        float2 U00 = make_float2(c, 0.f);
        float2 U01 = make_float2(-cosf(lam) * s, -sinf(lam) * s);
        float2 U10 = make_float2( cosf(ph)  * s,  sinf(ph)  * s);
        float2 U11 = make_float2( cosf(ph + lam) * c, sinf(ph + lam) * c);
        if (d.type == 0){  // I (x) U   (1q gate on low bit wb)
            M[0*4+0]=U00; M[0*4+1]=U01; M[1*4+0]=U10; M[1*4+1]=U11;
            M[2*4+2]=U00; M[2*4+3]=U01; M[3*4+2]=U10; M[3*4+3]=U11;
        } else {           // CU3: ctrl = high bit
            M[0*4+0]=make_float2(1.f,0.f); M[1*4+1]=make_float2(1.f,0.f);
            M[2*4+2]=U00; M[2*4+3]=U01; M[3*4+2]=U10; M[3*4+3]=U11;
        }
    } else if (d.type <= 3){
        float phi = conv[d.poff];
        float c = cosf(phi * 0.5f), s = sinf(phi * 0.5f);
        if (d.type == 1){          // exp(-i phi/2 ZZ)
            M[0]=make_float2(c,-s); M[5]=make_float2(c,s); M[10]=make_float2(c,s); M[15]=make_float2(c,-s);
        } else if (d.type == 2){   // exp(-i phi/2 YY)
            M[0]=M[5]=M[10]=M[15]=make_float2(c,0.f);
            M[0*4+3]=make_float2(0.f, s); M[1*4+2]=make_float2(0.f,-s);
            M[2*4+1]=make_float2(0.f,-s); M[3*4+0]=make_float2(0.f, s);
        } else {                   // exp(-i phi/2 XX)
            M[0]=M[5]=M[10]=M[15]=make_float2(c,0.f);
            M[0*4+3]=M[1*4+2]=M[2*4+1]=M[3*4+0]=make_float2(0.f,-s);
        }
    } else {                       // Pauli rotation: c I - i s (Pa (x) Pb)
        float th = last[d.poff];
        float c = cosf(th * 0.5f), s = sinf(th * 0.5f);
        int pa = d.aux >> 2, pb = d.aux & 3;
        for (int r = 0; r < 4; ++r){
            for (int cc = 0; cc < 4; ++cc){
                float2 pe = cmul(pel(pa, r >> 1, cc >> 1), pel(pb, r & 1, cc & 1));
                float2 v  = make_float2(s * pe.y, -s * pe.x);   // -i*s*pe
                if (r == cc) v.x += c;
                M[r * 4 + cc] = v;
            }
        }
    }
#pragma unroll
    for (int e = 0; e < 16; ++e) gws[g * 16 + e] = M[e];
}

// ---------- kernel C: evolve 256 basis columns through the 66-gate circuit ----------
__global__ void __launch_bounds__(256) sim_columns(const float2* __restrict__ gws,
                                                   float2* __restrict__ Ucols){
    __shared__ float2 st[16 * 256];                 // 16 columns, 32 KB
    int tid = threadIdx.x;
    int colBase = blockIdx.x * 16;
    for (int e = tid; e < 16 * 256; e += 256){
        int col = e >> 8, amp = e & 255;
        float2 v = make_float2(0.f, 0.f);
        if (amp == colBase + col) v.x = 1.f;
        st[e] = v;
    }
    __syncthreads();
    for (int g = 0; g < 66; ++g){
        GD d = g_gates[g];
        float2 M[16];
#pragma unroll
        for (int e = 0; e < 16; ++e) M[e] = gws[g * 16 + e];
        int ba = 7 - d.wa, bb = 7 - d.wb;           // wire w <-> bit (7-w)
        int lo = (ba < bb) ? ba : bb;
        int hb = (ba < bb) ? bb : ba;
        unsigned mlo = (1u << lo) - 1u, mhi = (1u << hb) - 1u;
        for (int task = tid; task < 16 * 64; task += 256){
            int col = task >> 6;
            unsigned t = (unsigned)(task & 63);
            unsigned u    = ((t & ~mlo) << 1) | (t & mlo);
            unsigned base = ((u & ~mhi) << 1) | (u & mhi);
            unsigned i0 = base;
            unsigned i1 = base | (1u << bb);
            unsigned i2 = base | (1u << ba);
            unsigned i3 = i1   | (1u << ba);
            float2* S = &st[col * 256];
            float2 av[4] = { S[i0], S[i1], S[i2], S[i3] };
            float2 rv[4];
#pragma unroll
            for (int r = 0; r < 4; ++r){
                float2 acc = make_float2(0.f, 0.f);
#pragma unroll
                for (int h = 0; h < 4; ++h) acc = cadd(acc, cmul(M[r * 4 + h], av[h]));
                rv[r] = acc;
            }
            S[i0] = rv[0]; S[i1] = rv[1]; S[i2] = rv[2]; S[i3] = rv[3];
        }
        __syncthreads();
    }
    for (int e = tid; e < 16 * 256; e += 256){
        int col = e >> 8, amp = e & 255;
        Ucols[(size_t)(colBase + col) * 256 + amp] = st[e];
    }
}

// ---------- kernel D: A = Re(U^dag Z0 U)  (real symmetric 256x256) ----------
__global__ void __launch_bounds__(256) build_A(const float2* __restrict__ Ucols,
                                               float* __restrict__ A){
    __shared__ float2 TJ[16][32];
    __shared__ float2 TK[16][32];
    int tx = threadIdx.x & 15, ty = threadIdx.x >> 4;
    int j0 = blockIdx.y * 16, k0 = blockIdx.x * 16;
    float acc = 0.f;
    for (int i0 = 0; i0 < 256; i0 += 32){
#pragma unroll
        for (int r = 0; r < 2; ++r){
            int e = threadIdx.x + r * 256;
            int jj = e >> 5, ii = e & 31;
            TJ[jj][ii] = Ucols[(size_t)(j0 + jj) * 256 + i0 + ii];
            TK[jj][ii] = Ucols[(size_t)(k0 + jj) * 256 + i0 + ii];
        }
        __syncthreads();
#pragma unroll
        for (int ii = 0; ii < 32; ++ii){
            float sgn = ((i0 + ii) & 128) ? -1.f : 1.f;   // Z on qubit 0 (MSB)
            float2 a = TJ[ty][ii], b = TK[tx][ii];
            acc += sgn * (a.x * b.x + a.y * b.y);
        }
        __syncthreads();
    }
    A[(size_t)(j0 + ty) * 256 + (k0 + tx)] = acc;
}

// ---------- kernel E: z0_b = v_b^T A v_b via WMMA; logits = z0*w_out + b_out ----------
typedef __attribute__((ext_vector_type(2))) float v2f;
typedef __attribute__((ext_vector_type(8))) float v8f;

__global__ void __launch_bounds__(64) quad_form(const float2* __restrict__ cs,
                                                const float* __restrict__ A,
                                                const float* __restrict__ ow,
                                                const float* __restrict__ ob,
                                                float* __restrict__ out){
    __shared__ float vhi[2][16][16];
    __shared__ float vlo[2][16][16];
    int w    = threadIdx.x >> 5;                    // wave index in block
    int lane = threadIdx.x & 31;
    int m0   = (blockIdx.x * 2 + w) * 16;           // 16 batch rows per wave

    // Build half-products of the RY product state:
    // v[m][i] = vhi[m][i>>4] * vlo[m][i&15],  wire w <-> bit (7-w)
    for (int t = lane; t < 256; t += 32){
        int m = t >> 4, h = t & 15;
        const float2* C = cs + (size_t)(m0 + m) * 8;
        float2 c0 = C[0], c1 = C[1], c2 = C[2], c3 = C[3];
        float2 c4 = C[4], c5 = C[5], c6 = C[6], c7 = C[7];
        float ph = ((h >> 3) & 1 ? c0.y : c0.x);
        ph      *= ((h >> 2) & 1 ? c1.y : c1.x);
        ph      *= ((h >> 1) & 1 ? c2.y : c2.x);
        ph      *= ((h     ) & 1 ? c3.y : c3.x);
        vhi[w][m][h] = ph;
        float pl = ((h >> 3) & 1 ? c4.y : c4.x);
        pl      *= ((h >> 2) & 1 ? c5.y : c5.x);
        pl      *= ((h >> 1) & 1 ? c6.y : c6.x);
        pl      *= ((h     ) & 1 ? c7.y : c7.x);
        vlo[w][m][h] = pl;
    }
    __syncthreads();

    int hi = lane >> 4, n = lane & 15;
    float q[8];
#pragma unroll
    for (int r = 0; r < 8; ++r) q[r] = 0.f;

#if __has_builtin(__builtin_amdgcn_wmma_f32_16x16x4_f32)
    for (int kt = 0; kt < 16; ++kt){
        v8f acc = {};
        int k = kt * 16 + n;                        // output column within A
        const float* Arow = A + (size_t)k * 256;    // A symmetric: A[j][k] == A[k][j]
        for (int js = 0; js < 64; ++js){
            int ja = js * 4 + 2 * hi;               // K-pair this half-wave supplies
            int jh = ja >> 4, jl = ja & 15;
            float vh = vhi[w][n][jh];
            v2f a, b;
            a.x = vh * vlo[w][n][jl];               // V[m=n][ja]
            a.y = vh * vlo[w][n][jl + 1];           // V[m=n][ja+1]  (jl <= 14 always)
            float2 bp = *(const float2*)(Arow + ja);
            b.x = bp.x;                             // A[ja][k]
            b.y = bp.y;                             // A[ja+1][k]
            acc = __builtin_amdgcn_wmma_f32_16x16x4_f32(
                      false, a, false, b, (short)0, acc, false, false);
        }
        // acc[r] = P[m][k] with m = r + 8*hi, k = kt*16 + n ; q += P .* V
#pragma unroll
        for (int r = 0; r < 8; ++r){
            int m = r + 8 * hi;
            q[r] += acc[r] * vhi[w][m][kt] * vlo[w][m][n];
        }
    }
#pragma unroll
    for (int r = 0; r < 8; ++r){
        q[r] += __shfl_xor(q[r], 1, 32);
        q[r] += __shfl_xor(q[r], 2, 32);
        q[r] += __shfl_xor(q[r], 4, 32);
        q[r] += __shfl_xor(q[r], 8, 32);            // masks < 16: stay within half-wave
    }
    if (n == 0){                                    // lanes 0 and 16 hold rows 0-7 / 8-15
#pragma unroll
        for (int r = 0; r < 8; ++r){
            int row = m0 + r + 8 * hi;
            float z = q[r];
            for (int cls = 0; cls < 10; ++cls)
                out[(size_t)row * 10 + cls] = z * ow[cls] + ob[cls];
        }
    }
#else
    // VALU fallback (no f32x4 WMMA builtin available)
    for (int mm = 0; mm < 16; ++mm){
        float qp = 0.f;
        for (int t = 0; t < 8; ++t){
            int k = lane + t * 32;
            float vk = vhi[w][mm][k >> 4] * vlo[w][mm][k & 15];
            float wk = 0.f;
            for (int j = 0; j < 256; ++j)
                wk += vhi[w][mm][j >> 4] * vlo[w][mm][j & 15] * A[(size_t)j * 256 + k];
            qp += vk * wk;
        }
        for (int m = 16; m >= 1; m >>= 1) qp += __shfl_xor(qp, m, 32);
        if (lane == 0){
            int row = m0 + mm;
            for (int cls = 0; cls < 10; ++cls)
                out[(size_t)row * 10 + cls] = qp * ow[cls] + ob[cls];
        }
    }
#endif
}

// ---------- host launcher ----------
extern "C" void kernel_launch(void* const* d_in, const int* in_sizes, int n_in,
                              void* d_out, int out_size, void* d_ws, size_t ws_size,
                              hipStream_t stream){
    const float* x    = (const float*)d_in[0];   // 16384 x 3072
    const float* fcw  = (const float*)d_in[1];   // 8 x 3072
    const float* fcb  = (const float*)d_in[2];   // 8
    const float* conv = (const float*)d_in[3];   // 2 x 8 x 15
    const float* pool = (const float*)d_in[4];   // 2 x 4 x 3
    const float* last = (const float*)d_in[5];   // 15
    const float* ow   = (const float*)d_in[6];   // 10
    const float* ob   = (const float*)d_in[7];   // 10
    float* out = (float*)d_out;                  // 16384 x 10

    char* ws = (char*)d_ws;
    float2* cs    = (float2*)(ws);                                   // 1 MB
    float2* gws   = (float2*)(ws + (1u << 20));                      // 16 KB reserved
    float2* Ucols = (float2*)(ws + (1u << 20) + (1u << 14));         // 512 KB
    float*  A     = (float*) (ws + (1u << 20) + (1u << 14) + (1u << 19)); // 256 KB

    fc_tanh    <<<2048, 256, 0, stream>>>(x, fcw, fcb, cs);
    build_gates<<<1,    96,  0, stream>>>(conv, pool, last, gws);
    sim_columns<<<16,   256, 0, stream>>>(gws, Ucols);
    build_A    <<<dim3(16, 16), 256, 0, stream>>>(Ucols, A);
    quad_form  <<<512,  64,  0, stream>>>(cs, A, ow, ob, out);
}